// HyvarinenLoss_37271726194741
// MI455X (gfx1250) — compile-verified
//
#include <hip/hip_runtime.h>

typedef float v2f __attribute__((ext_vector_type(2)));
typedef float v8f __attribute__((ext_vector_type(8)));

#define B_N 32768
#define D_N 64
#define H_N 256

// ---------------------------------------------------------------------------
// Kernel 1: c[k] = sum_i W1[i,k] * W2[k,i]   (analytic Jacobian-trace weights)
// ---------------------------------------------------------------------------
__global__ void hyva_precompute_c(const float* __restrict__ W1,
                                  const float* __restrict__ W2,
                                  float* __restrict__ cvec) {
    int k = threadIdx.x;            // 256 threads, one per hidden unit
    float acc = 0.0f;
#pragma unroll 8
    for (int i = 0; i < D_N; ++i)
        acc += W1[i * H_N + k] * W2[k * D_N + i];
    cvec[k] = acc;
}

// ---------------------------------------------------------------------------
// Kernel 2: pack W1 / W2 into WMMA B-fragment-major float2 arrays so the hot
// loop issues one coalesced global_load_b64 per fragment per lane.
//   W1 frag (nt in 0..15, ks in 0..15): lane l -> {W1[k0, col], W1[k0+1, col]}
//       k0 = ks*4 + 2*(l>>4), col = nt*16 + (l&15)
//   W2 frag (nt in 0..15, ks2 in 0..3, t in 0..3): lane l ->
//       {W2[kg, col], W2[kg+1, col]}, kg = nt*16+ks2*4+2*(l>>4), col = t*16+(l&15)
// ---------------------------------------------------------------------------
__global__ void hyva_pack_weights(const float* __restrict__ W1,
                                  const float* __restrict__ W2,
                                  v2f* __restrict__ pW1,
                                  v2f* __restrict__ pW2) {
    int idx = blockIdx.x * blockDim.x + threadIdx.x;   // 16384 threads
    int lane = idx & 31;
    int frag = (idx >> 5) & 255;
    int n    = lane & 15;
    int half = lane >> 4;
    if (idx < 8192) {                                  // W1: frag = nt*16 + ks
        int ks = frag & 15;
        int nt = frag >> 4;
        int k0 = ks * 4 + 2 * half;
        int col = nt * 16 + n;
        v2f o;
        o.x = W1[(k0 + 0) * H_N + col];
        o.y = W1[(k0 + 1) * H_N + col];
        pW1[idx] = o;
    } else {                                           // W2: frag = (nt*4+ks2)*4+t
        int t   = frag & 3;
        int ks2 = (frag >> 2) & 3;
        int nt  = frag >> 4;
        int kg  = nt * 16 + ks2 * 4 + 2 * half;
        int col = t * 16 + n;
        v2f o;
        o.x = W2[(kg + 0) * D_N + col];
        o.y = W2[(kg + 1) * D_N + col];
        pW2[idx - 8192] = o;
    }
}

// ---------------------------------------------------------------------------
// Kernel 3: fused  h = xW1+b1 ; a = tanh(h) ; s = aW2+b2 ;
//           loss += 0.5*|s|^2 + sum_k (1-a_k^2) c_k      (per-block partials)
// One wave32 handles a 16-row tile using V_WMMA_F32_16X16X4_F32.
// ---------------------------------------------------------------------------
__launch_bounds__(256)
__global__ void hyva_main(const float* __restrict__ x,
                          const v2f* __restrict__ pW1,
                          const v2f* __restrict__ pW2,
                          const float* __restrict__ b1,
                          const float* __restrict__ b2,
                          const float* __restrict__ cvec,
                          float* __restrict__ partials) {
    __shared__ float s_c[H_N];
    __shared__ float s_b1[H_N];
    __shared__ float s_b2[D_N];
    __shared__ float s_tile[8][16 * 16];   // per-wave transpose staging
    __shared__ float s_red[8];

    const int tid = threadIdx.x;
    s_c[tid]  = cvec[tid];
    s_b1[tid] = b1[tid];
    if (tid < D_N) s_b2[tid] = b2[tid];
    __syncthreads();

    const int wave = tid >> 5;
    const int lane = tid & 31;
    const int n    = lane & 15;     // N / M index within 16
    const int half = lane >> 4;     // selects K pair {0,1} vs {2,3}

    const int row0 = (blockIdx.x * 8 + wave) * 16;

    // Preload A fragments for GEMM1: lane holds row (row0+n), K = ks*4+2*half+{0,1}
    const float* xrow = x + (size_t)(row0 + n) * D_N + 2 * half;
    v2f xa[16];
#pragma unroll
    for (int ks = 0; ks < 16; ++ks)
        xa[ks] = *(const v2f*)(xrow + ks * 4);

    const v8f zero8 = {0.f, 0.f, 0.f, 0.f, 0.f, 0.f, 0.f, 0.f};
    v8f acc2[4];                    // S tile accumulators: 4 n-tiles of D=64
#pragma unroll
    for (int t = 0; t < 4; ++t) acc2[t] = zero8;

    float loss = 0.0f;
    float* tile = &s_tile[wave][0];
    const v2f* pW1l = pW1 + lane;   // fragment-major, lane-coalesced
    const v2f* pW2l = pW2 + lane;

    for (int nt = 0; nt < 16; ++nt) {        // stream 16 column-tiles of H
        // ---- GEMM1: h[16x16] = x_tile[16x64] @ W1[:, nt*16 : nt*16+16] ----
        // Two independent accumulator chains to overlap WMMA RAW latency.
        v8f h0 = zero8, h1 = zero8;
        const v2f* w1f = pW1l + (size_t)(nt * 16) * 32;
#pragma unroll
        for (int ks = 0; ks < 16; ks += 2) {
            v2f bfa = w1f[(ks + 0) * 32];
            v2f bfb = w1f[(ks + 1) * 32];
            h0 = __builtin_amdgcn_wmma_f32_16x16x4_f32(
                false, xa[ks + 0], false, bfa, (short)0, h0, false, false);
            h1 = __builtin_amdgcn_wmma_f32_16x16x4_f32(
                false, xa[ks + 1], false, bfb, (short)0, h1, false, false);
        }

        // ---- bias + tanh + trace term; stash activations to LDS ----
        const int   hcol = nt * 16 + n;
        const float bias = s_b1[hcol];
        const float cv   = s_c[hcol];
#pragma unroll
        for (int v = 0; v < 8; ++v) {
            float a = tanhf(h0[v] + h1[v] + bias);
            loss += (1.0f - a * a) * cv;                 // analytic trace
            tile[(v + 8 * half) * 16 + n] = a;           // C-layout -> (m,n)
        }
        asm volatile("s_wait_dscnt 0" ::: "memory");     // stores visible wave-wide

        // ---- GEMM2 partial: S += a_tile[16x16] @ W2[nt*16 : nt*16+16, :] ----
        const v2f* w2f = pW2l + (size_t)(nt * 4) * 4 * 32;
#pragma unroll
        for (int ks2 = 0; ks2 < 4; ++ks2) {
            const int kk = ks2 * 4 + 2 * half;
            v2f af = *(const v2f*)(tile + n * 16 + kk);  // A 16x4 fragment (ds b64)
#pragma unroll
            for (int t = 0; t < 4; ++t) {
                v2f bf = w2f[(ks2 * 4 + t) * 32];
                acc2[t] = __builtin_amdgcn_wmma_f32_16x16x4_f32(
                    false, af, false, bf, (short)0, acc2[t], false, false);
            }
        }
    }

    // ---- s = acc2 + b2 ; loss += 0.5*s^2 ----
#pragma unroll
    for (int t = 0; t < 4; ++t) {
        const float bias2 = s_b2[t * 16 + n];
#pragma unroll
        for (int v = 0; v < 8; ++v) {
            float s = acc2[t][v] + bias2;
            loss += 0.5f * s * s;
        }
    }

    // ---- deterministic reduction: wave shfl tree -> LDS -> block partial ----
#pragma unroll
    for (int off = 16; off > 0; off >>= 1)
        loss += __shfl_xor(loss, off, 32);
    if (lane == 0) s_red[wave] = loss;
    __syncthreads();
    if (tid == 0) {
        float t = 0.0f;
#pragma unroll
        for (int w = 0; w < 8; ++w) t += s_red[w];
        partials[blockIdx.x] = t;
    }
}

// ---------------------------------------------------------------------------
// Kernel 4: deterministic final reduce of 256 block partials -> mean loss
// ---------------------------------------------------------------------------
__global__ void hyva_reduce(const float* __restrict__ partials,
                            float* __restrict__ out) {
    __shared__ float red[256];
    int t = threadIdx.x;
    red[t] = partials[t];
    __syncthreads();
    for (int s = 128; s > 0; s >>= 1) {
        if (t < s) red[t] += red[t + s];
        __syncthreads();
    }
    if (t == 0) out[0] = red[0] * (1.0f / (float)B_N);
}

extern "C" void kernel_launch(void* const* d_in, const int* in_sizes, int n_in,
                              void* d_out, int out_size, void* d_ws, size_t ws_size,
                              hipStream_t stream) {
    const float* x  = (const float*)d_in[0];   // [32768, 64]
    const float* W1 = (const float*)d_in[1];   // [64, 256]
    const float* b1 = (const float*)d_in[2];   // [256]
    const float* W2 = (const float*)d_in[3];   // [256, 64]
    const float* b2 = (const float*)d_in[4];   // [64]

    float* ws       = (float*)d_ws;
    float* cvec     = ws;                   // 256 floats
    float* partials = ws + 256;             // 256 floats
    v2f*   pW1      = (v2f*)(ws + 512);     // 8192 float2  (64 KB)
    v2f*   pW2      = pW1 + 8192;           // 8192 float2  (64 KB)

    hyva_precompute_c<<<1, 256, 0, stream>>>(W1, W2, cvec);
    hyva_pack_weights<<<64, 256, 0, stream>>>(W1, W2, pW1, pW2);
    hyva_main<<<256, 256, 0, stream>>>(x, pW1, pW2, b1, b2, cvec, partials);
    hyva_reduce<<<1, 256, 0, stream>>>(partials, (float*)d_out);
}